// DiGCN_IB_1BN_batch_46746424050290
// MI455X (gfx1250) — compile-verified
//
#include <hip/hip_runtime.h>

typedef __attribute__((ext_vector_type(2))) float v2f;
typedef __attribute__((ext_vector_type(8))) float v8f;

#define FD 128            // feature dim (K)
#define CD 64             // classes per matrix
#define NC 192            // combined output columns (Wl|W1|W2)
#define ROWS_PER_BLOCK 64 // 4 waves x 16-row strips
#define BN_EPS 1e-5f

// ---------------------------------------------------------------------------
// Transpose [Wl|W1|W2] (each row-major [128x64]) into Wt[ct*128 + k]
// (col-major combined 192x128) so WMMA B fragments are contiguous b64 loads.
// ---------------------------------------------------------------------------
__global__ void prep_weights(const float* __restrict__ Wl,
                             const float* __restrict__ W1,
                             const float* __restrict__ W2,
                             float* __restrict__ Wt) {
  int idx = blockIdx.x * blockDim.x + threadIdx.x;
  if (idx >= NC * FD) return;
  int ct = idx / FD;  // combined column 0..191
  int k  = idx % FD;  // 0..127
  const float* Wsrc = (ct < CD) ? Wl : ((ct < 2 * CD) ? W1 : W2);
  int c = ct & (CD - 1);
  Wt[idx] = Wsrc[k * CD + c];
}

// ---------------------------------------------------------------------------
// Fused triple GEMM: each block stages 64 rows of x in LDS, each wave (32
// lanes) computes a 16-row strip against all 192 columns with
// V_WMMA_F32_16X16X4_F32, accumulating K=128 in steps of 4.
//   h0 -> acc buffer (overwrite), h1/h2 -> workspace for edge gather.
// ---------------------------------------------------------------------------
__global__ void __launch_bounds__(128)
fused_gemm(const float* __restrict__ x, const float* __restrict__ Wt,
           float* __restrict__ h0, float* __restrict__ h1,
           float* __restrict__ h2, int nrows) {
  __shared__ float xs[ROWS_PER_BLOCK * FD];  // 32 KB

  int tid = threadIdx.x;
  long base = (long)blockIdx.x * ROWS_PER_BLOCK;

  // Cooperative, coalesced stage of 64x128 floats (2048 float4).
  const int NV4 = ROWS_PER_BLOCK * FD / 4;
  for (int i = tid; i < NV4; i += 128) {
    int r  = i >> 5;  // 32 float4 per row
    int c4 = i & 31;
    long gr = base + r;
    if (gr >= nrows) gr = nrows - 1;  // clamp tail (stores masked later)
    float4 v = reinterpret_cast<const float4*>(x + gr * FD)[c4];
    reinterpret_cast<float4*>(xs)[i] = v;
  }
  __syncthreads();

  int wave  = tid >> 5;
  int lane  = tid & 31;
  int mrow  = lane & 15;   // row within 16-tile / col within 16-tile
  int khalf = lane >> 4;   // upper-lane half handles K=+2,+3

  v8f accv[12] = {};

  const float* arow = xs + (wave * 16 + mrow) * FD;

  for (int k0 = 0; k0 < FD; k0 += 4) {
    int k = k0 + khalf * 2;
    // A fragment (16x4 f32): lane<16 -> K=k0,k0+1 ; lane>=16 -> K=k0+2,k0+3
    v2f a = *reinterpret_cast<const v2f*>(arow + k);
#pragma unroll
    for (int t = 0; t < 12; ++t) {
      // B fragment (4x16 f32), col = t*16 + mrow, col-major Wt -> b64 load
      v2f b = *reinterpret_cast<const v2f*>(Wt + (t * 16 + mrow) * FD + k);
      accv[t] = __builtin_amdgcn_wmma_f32_16x16x4_f32(
          false, a, false, b, (short)0, accv[t], false, false);
    }
  }

  // C/D layout: VGPR i holds M = i (lanes 0-15) or M = i+8 (lanes 16-31),
  // N = lane % 16.
  float* dests[3] = {h0, h1, h2};
  int roff = (lane >> 4) << 3;
#pragma unroll
  for (int t = 0; t < 12; ++t) {
    float* d = dests[t >> 2];
    int colbase = (t & 3) * 16;
#pragma unroll
    for (int i = 0; i < 8; ++i) {
      long r = base + wave * 16 + i + roff;
      if (r < nrows) d[r * CD + colbase + mrow] = accv[t][i];
    }
  }
}

// ---------------------------------------------------------------------------
// Edge scatter: thread-per-edge; ~99% of edges fail the same-block test and
// exit immediately (pure streaming read of the edge list). Survivors gather
// 64 floats of h[src] and atomically add w-scaled values into acc[dst].
// ---------------------------------------------------------------------------
__global__ void edge_scatter(const int* __restrict__ ei,
                             const float* __restrict__ ew,
                             const float* __restrict__ h,
                             float* __restrict__ acc, int E) {
  int e = blockIdx.x * blockDim.x + threadIdx.x;
  if (e >= E) return;
  int s = ei[e];
  int d = ei[E + e];
  if ((s >> 10) != (d >> 10)) return;  // BS = 1024
  float w = ew[e];
  const float4* hp = reinterpret_cast<const float4*>(h + (long)s * CD);
  float* ap = acc + (long)d * CD;
#pragma unroll
  for (int i = 0; i < CD / 4; ++i) {
    float4 v = hp[i];
    atomicAdd(ap + i * 4 + 0, v.x * w);
    atomicAdd(ap + i * 4 + 1, v.y * w);
    atomicAdd(ap + i * 4 + 2, v.z * w);
    atomicAdd(ap + i * 4 + 3, v.w * w);
  }
}

// ---------------------------------------------------------------------------
// Epilogue: fold the three biases + eval-mode BatchNorm affine.
// ---------------------------------------------------------------------------
__global__ void bn_finalize(const float* __restrict__ acc,
                            const float* __restrict__ bl,
                            const float* __restrict__ b1,
                            const float* __restrict__ b2,
                            const float* __restrict__ gamma,
                            const float* __restrict__ beta,
                            const float* __restrict__ rmean,
                            const float* __restrict__ rvar,
                            float* __restrict__ out, long total) {
  long idx = (long)blockIdx.x * blockDim.x + threadIdx.x;
  if (idx >= total) return;
  int c = (int)(idx & (CD - 1));
  float inv = gamma[c] * rsqrtf(rvar[c] + BN_EPS);
  out[idx] = (acc[idx] + bl[c] + b1[c] + b2[c] - rmean[c]) * inv + beta[c];
}

// ---------------------------------------------------------------------------
extern "C" void kernel_launch(void* const* d_in, const int* in_sizes, int n_in,
                              void* d_out, int out_size, void* d_ws,
                              size_t ws_size, hipStream_t stream) {
  const float* x     = (const float*)d_in[0];
  const int*   ei1   = (const int*)d_in[1];
  const float* ew1   = (const float*)d_in[2];
  const int*   ei2   = (const int*)d_in[3];
  const float* ew2   = (const float*)d_in[4];
  const float* Wl    = (const float*)d_in[5];
  const float* bl    = (const float*)d_in[6];
  const float* W1    = (const float*)d_in[7];
  const float* b1    = (const float*)d_in[8];
  const float* W2    = (const float*)d_in[9];
  const float* b2    = (const float*)d_in[10];
  const float* gamma = (const float*)d_in[11];
  const float* beta  = (const float*)d_in[12];
  const float* rmean = (const float*)d_in[13];
  const float* rvar  = (const float*)d_in[14];

  int n  = in_sizes[0] / FD;  // 100000 nodes
  int E1 = in_sizes[2];
  int E2 = in_sizes[4];

  // Workspace layout (floats): Wt | acc | h1 | h2  (~77 MB total)
  float* wsf = (float*)d_ws;
  float* Wt  = wsf;
  float* acc = wsf + (size_t)NC * FD;
  float* h1  = acc + (size_t)n * CD;
  float* h2  = h1 + (size_t)n * CD;
  float* out = (float*)d_out;

  prep_weights<<<(NC * FD + 255) / 256, 256, 0, stream>>>(Wl, W1, W2, Wt);

  fused_gemm<<<(n + ROWS_PER_BLOCK - 1) / ROWS_PER_BLOCK, 128, 0, stream>>>(
      x, Wt, acc, h1, h2, n);

  edge_scatter<<<(E1 + 255) / 256, 256, 0, stream>>>(ei1, ew1, h1, acc, E1);
  edge_scatter<<<(E2 + 255) / 256, 256, 0, stream>>>(ei2, ew2, h2, acc, E2);

  long total = (long)n * CD;
  bn_finalize<<<(int)((total + 255) / 256), 256, 0, stream>>>(
      acc, bl, b1, b2, gamma, beta, rmean, rvar, out, total);
}